// GradientNormLayer_85212151153362
// MI455X (gfx1250) — compile-verified
//
#include <hip/hip_runtime.h>
#include <math.h>

typedef float v2f __attribute__((ext_vector_type(2)));
typedef float v8f __attribute__((ext_vector_type(8)));

#define D_DIM 256
#define C_DIM 256

// One block = 16 waves = 512 threads; computes 16 rows of the batch.
// Phase 1: stage 16x256 h-tile in LDS.
// Phase 2: logits(16x256) = h_tile @ W + b via v_wmma_f32_16x16x4_f32
//          (wave w owns columns [16w, 16w+16), K-loop of 64 steps).
// Phase 3: wave w reduces row w: softmax stats + ||h_row||^2 ->
//          out = sqrt((||h||^2+1) * (sum p^2 - 2 p_label + 1)).
__global__ __launch_bounds__(512) void gradnorm_fused_kernel(
    const float* __restrict__ h, const float* __restrict__ W,
    const float* __restrict__ bias, const int* __restrict__ labels,
    float* __restrict__ out)
{
    __shared__ float lds_h[16 * D_DIM];       // 16 KB
    __shared__ float lds_logits[16 * C_DIM];  // 16 KB

    const int tid  = threadIdx.x;   // 0..511
    const int wave = tid >> 5;      // 0..15
    const int lane = tid & 31;      // 0..31
    const int row0 = blockIdx.x * 16;

    // ---- Phase 1: coalesced h tile -> LDS (4096 floats / 512 threads) ----
#pragma unroll
    for (int i = 0; i < 8; ++i) {
        const int idx = tid + i * 512;
        lds_h[idx] = h[row0 * D_DIM + idx];
    }
    __syncthreads();

    // ---- Phase 2: WMMA GEMM, f32 16x16x4 ----
    const int hi   = lane >> 4;          // lane-half: selects K pair {0,1} vs {2,3}
    const int mrow = lane & 15;          // A-matrix M index / D-matrix N index
    const int ncol = wave * 16 + mrow;   // this wave's logit column for B/D

    v8f acc = {};
    for (int k0 = 0; k0 < D_DIM; k0 += 4) {
        // A (16x4): lane m=mrow holds K = {2*hi, 2*hi+1} in vgprs {0,1}
        const v2f a = *(const v2f*)&lds_h[mrow * D_DIM + k0 + 2 * hi];
        // B (4x16): lane n=mrow, lane-half selects K rows {2*hi, 2*hi+1}
        v2f bm;
        bm.x = W[(k0 + 2 * hi + 0) * C_DIM + ncol];
        bm.y = W[(k0 + 2 * hi + 1) * C_DIM + ncol];
        acc = __builtin_amdgcn_wmma_f32_16x16x4_f32(
            /*neg_a=*/false, a, /*neg_b=*/false, bm,
            /*c_mod=*/(short)0, acc, /*reuse_a=*/false, /*reuse_b=*/false);
    }

    // D layout: acc[j] -> row (j + 8*hi), col ncol. Add bias, spill to LDS.
    const float bcol = bias[ncol];
#pragma unroll
    for (int j = 0; j < 8; ++j) {
        lds_logits[(j + 8 * hi) * C_DIM + ncol] = acc[j] + bcol;
    }
    __syncthreads();

    // ---- Phase 3: wave `wave` reduces row `wave` ----
    const int row  = wave;
    const int grow = row0 + row;

    float vals[8];
    float lmax = -INFINITY;
#pragma unroll
    for (int i = 0; i < 8; ++i) {
        vals[i] = lds_logits[row * C_DIM + lane + 32 * i];
        lmax = fmaxf(lmax, vals[i]);
    }
#pragma unroll
    for (int off = 16; off > 0; off >>= 1)
        lmax = fmaxf(lmax, __shfl_xor(lmax, off, 32));

    float lsum = 0.f;
#pragma unroll
    for (int i = 0; i < 8; ++i) { vals[i] = __expf(vals[i] - lmax); lsum += vals[i]; }
#pragma unroll
    for (int off = 16; off > 0; off >>= 1)
        lsum += __shfl_xor(lsum, off, 32);

    const float inv = 1.0f / lsum;
    const int   lbl = labels[grow];
    float psq = 0.f, plbl = 0.f;
#pragma unroll
    for (int i = 0; i < 8; ++i) {
        const float p = vals[i] * inv;
        psq += p * p;
        if ((lane + 32 * i) == lbl) plbl = p;
    }

    float hn2 = 0.f;
#pragma unroll
    for (int i = 0; i < 8; ++i) {
        const float hv = lds_h[row * D_DIM + lane + 32 * i];
        hn2 += hv * hv;
    }
#pragma unroll
    for (int off = 16; off > 0; off >>= 1) {
        psq  += __shfl_xor(psq,  off, 32);
        plbl += __shfl_xor(plbl, off, 32);
        hn2  += __shfl_xor(hn2,  off, 32);
    }

    if (lane == 0) {
        const float g2 = psq - 2.f * plbl + 1.f;   // ||softmax - onehot||^2
        out[grow] = sqrtf((hn2 + 1.f) * g2);
    }
}

extern "C" void kernel_launch(void* const* d_in, const int* in_sizes, int n_in,
                              void* d_out, int out_size, void* d_ws, size_t ws_size,
                              hipStream_t stream) {
    const float* h      = (const float*)d_in[0];  // [B, 256]
    const float* W      = (const float*)d_in[1];  // [256, 256]
    const float* bias   = (const float*)d_in[2];  // [256]
    const int*   labels = (const int*)d_in[3];    // [B]
    float*       out    = (float*)d_out;          // [B, 1]

    const int B = in_sizes[3];        // 2048
    const int blocks = B / 16;        // 128
    gradnorm_fused_kernel<<<blocks, 512, 0, stream>>>(h, W, bias, labels, out);
}